// Cochlear_22978075033866
// MI455X (gfx1250) — compile-verified
//
#include <hip/hip_runtime.h>

// ---------------------------------------------------------------------------
// Gammatone filterbank (4 cascaded complex one-pole IIRs) on MI455X/gfx1250.
//
// Block-parallel linear recurrence, L=16 == WMMA M:
//   phase1 : beta = B_c * U                       (WMMA f32 16x16x4)
//   phase2 : hierarchical affine scan over blocks (3-pass, closed-form jumps)
//   phase3 : out = H_c*U + PsiR*sig_r + PsiN*sig_i (WMMA f32 16x16x4)
// Workspace: mats (~352KB) + S (62.5 MiB) + tau (512 KB).
// ---------------------------------------------------------------------------

typedef __attribute__((ext_vector_type(2))) float v2f;
typedef __attribute__((ext_vector_type(8))) float v8f;

#define BATCH 8
#define TLEN  32000
#define CH    128
#define LBLK  16                 // samples per block == WMMA M
#define NBLK  (TLEN / LBLK)      // 2000 blocks per (b,c)
#define TILE  16                 // blocks per WMMA tile == WMMA N
#define NTILE (NBLK / TILE)      // 125 tiles per (b,c)
#define SEG   16                 // scan segments per (b,c)
#define SEGBLK (NBLK / SEG)      // 125 blocks per segment

struct ChanMats {
    float Hr[16][16];   // Re of 4-stage impulse-response Toeplitz, factor folded
    float Bs[16][16];   // rows 0..3 Re(B_m), 4..7 Im(B_m), 8..15 zero; factor folded
    float PsiR[16][4];  // Re of entry-state->output map
    float PsiN[16][4];  // -Im of entry-state->output map
    float Are[4][4];    // per-block (16-sample) state transition, lower tri
    float Aim[4][4];
    float Sre[4][4];    // per-segment (2000-sample) state transition, lower tri
    float Sim[4][4];
};

// C(n+d, d) for d in 0..3 (works for large n in f32 to ~1e-7 rel err)
__device__ __host__ inline float binc(int d, int n) {
    float nf = (float)n;
    if (d == 0) return 1.f;
    if (d == 1) return nf + 1.f;
    if (d == 2) return (nf + 1.f) * (nf + 2.f) * 0.5f;
    return (nf + 1.f) * (nf + 2.f) * (nf + 3.f) * (1.f / 6.f);
}

__device__ inline v8f wmma4(v2f a, v2f b, v8f c) {
    // D(16x16) = A(16x4) * B(4x16) + C   -- v_wmma_f32_16x16x4_f32
    return __builtin_amdgcn_wmma_f32_16x16x4_f32(
        false, a, false, b, (short)0, c, false, false);
}

// ------------------------------ K0: per-channel matrices --------------------
__global__ void precompute_mats(const float* __restrict__ coef_re,
                                const float* __restrict__ coef_im,
                                const float* __restrict__ factor,
                                ChanMats* __restrict__ mats) {
    int c = blockIdx.x * blockDim.x + threadIdx.x;
    if (c >= CH) return;
    float ar = coef_re[c], ai = coef_im[c], f = factor[c];
    float pr[17], pi[17];
    pr[0] = 1.f; pi[0] = 0.f;
    for (int n = 1; n <= 16; ++n) {        // complex powers a^n
        pr[n] = pr[n-1] * ar - pi[n-1] * ai;
        pi[n] = pr[n-1] * ai + pi[n-1] * ar;
    }
    ChanMats& M = mats[c];
    // H: y_zs[t] = sum_k C(t-k+3,3) a^(t-k) * (f * inp[k]); real part only.
    for (int t = 0; t < 16; ++t)
        for (int k = 0; k < 16; ++k)
            M.Hr[t][k] = (t >= k) ? f * binc(3, t - k) * pr[t - k] : 0.f;
    // B: zero-state final state m = m-stage cascade impulse at delay 15-k.
    for (int m = 0; m < 16; ++m)
        for (int k = 0; k < 16; ++k) M.Bs[m][k] = 0.f;
    for (int m0 = 0; m0 < 4; ++m0)
        for (int k = 0; k < 16; ++k) {
            int d = 15 - k;
            float w = f * binc(m0, d);
            M.Bs[m0][k]     = w * pr[d];
            M.Bs[m0 + 4][k] = w * pi[d];
        }
    // Psi: entry state j contributes C(t+4-j,4-j) a^(t+1) to s4[t].
    for (int t = 0; t < 16; ++t)
        for (int j0 = 0; j0 < 4; ++j0) {
            float w = binc(3 - j0, t);
            M.PsiR[t][j0] =  w * pr[t + 1];
            M.PsiN[t][j0] = -w * pi[t + 1];   // minus folded: Re(z*s)=Rr*sr - Ri*si
        }
    // Per-block transition: T(16)[m][j] = C(15+m-j, m-j) a^16.
    for (int m = 0; m < 4; ++m)
        for (int j = 0; j < 4; ++j) {
            if (j <= m) {
                float w = binc(m - j, 15);
                M.Are[m][j] = w * pr[16];
                M.Aim[m][j] = w * pi[16];
            } else { M.Are[m][j] = 0.f; M.Aim[m][j] = 0.f; }
        }
    // Per-segment transition: T(2000)[m][j] = C(1999+m-j, m-j) a^2000,
    // a^2000 = (a^16)^125 by binary exponentiation.
    {
        float br = pr[16], bi2 = pi[16];      // base = a^16
        float rr = 1.f, ri = 0.f;             // result
        int e = SEGBLK;                       // 125
        while (e) {
            if (e & 1) {
                float tr = rr * br - ri * bi2;
                ri = rr * bi2 + ri * br;
                rr = tr;
            }
            e >>= 1;
            if (e) {
                float tr = br * br - bi2 * bi2;
                bi2 = 2.f * br * bi2;
                br = tr;
            }
        }
        for (int m = 0; m < 4; ++m)
            for (int j = 0; j < 4; ++j) {
                if (j <= m) {
                    float w = binc(m - j, LBLK * SEGBLK - 1);
                    M.Sre[m][j] = w * rr;
                    M.Sim[m][j] = w * ri;
                } else { M.Sre[m][j] = 0.f; M.Sim[m][j] = 0.f; }
            }
    }
}

// ------------------------------ K1: beta = Bs * U (WMMA) --------------------
// One wave per (channel, batch). A-frag: lane%16 = row M, {lane/16, vgpr} = K.
// B-frag (U tile: K=time-in-block, N=block): lane%16 = N, {lane/16, vgpr} = K.
// D: vgpr v + 8*(lane/16) = row M, lane%16 = col N.
__global__ void __launch_bounds__(32)
phase1_states(const float* __restrict__ inp, const ChanMats* __restrict__ mats,
              float* __restrict__ S) {
    const int bid  = blockIdx.x;
    const int c    = bid & (CH - 1);
    const int b    = bid >> 7;
    const int lane = threadIdx.x;
    const int half = lane >> 4;
    const int lm   = lane & 15;
    const ChanMats& M = mats[c];

    v2f aB[4];
    #pragma unroll
    for (int q = 0; q < 4; ++q)
        aB[q] = *(const v2f*)&M.Bs[lm][4 * q + 2 * half];

    const float* x = inp + (size_t)b * TLEN;
    float* Srow    = S + (size_t)(c * BATCH + b) * NBLK * 8;

    for (int tile = 0; tile < NTILE; ++tile) {
        const float* xt = x + tile * (LBLK * TILE);
        v8f acc = {};
        #pragma unroll
        for (int q = 0; q < 4; ++q) {
            v2f bu = *(const v2f*)&xt[lm * LBLK + 4 * q + 2 * half];
            acc = wmma4(aB[q], bu, acc);
        }
        // Rows 0..7 = [Re s1..s4, Im s1..s4]; rows 8..15 zero by construction.
        if (half == 0) {
            float* dst = Srow + (size_t)(tile * TILE + lm) * 8;
            #pragma unroll
            for (int v = 0; v < 8; ++v) dst[v] = acc[v];
        }
    }
}

// -------------------- shared affine-scan body (4x4 complex, lower tri) ------
template <bool WRITE_SIGMA>
__device__ inline void scan_blocks(float* p, int nblk,
                                   const float Are[4][4], const float Aim[4][4],
                                   float sr[4], float si[4]) {
    for (int n = 0; n < nblk; ++n, p += 8) {
        float br[4], bi[4];
        #pragma unroll
        for (int j = 0; j < 4; ++j) { br[j] = p[j]; bi[j] = p[4 + j]; }
        if (WRITE_SIGMA) {
            #pragma unroll
            for (int j = 0; j < 4; ++j) { p[j] = sr[j]; p[4 + j] = si[j]; }
        }
        float nr[4], ni[4];
        #pragma unroll
        for (int m = 0; m < 4; ++m) {
            float xr = br[m], xi = bi[m];
            #pragma unroll
            for (int j = 0; j < 4; ++j) {
                if (j <= m) {
                    xr += Are[m][j] * sr[j] - Aim[m][j] * si[j];
                    xi += Are[m][j] * si[j] + Aim[m][j] * sr[j];
                }
            }
            nr[m] = xr; ni[m] = xi;
        }
        #pragma unroll
        for (int m = 0; m < 4; ++m) { sr[m] = nr[m]; si[m] = ni[m]; }
    }
}

__device__ inline void load44(const float src[4][4], float dst[4][4]) {
    #pragma unroll
    for (int m = 0; m < 4; ++m)
        #pragma unroll
        for (int j = 0; j < 4; ++j) dst[m][j] = src[m][j];
}

// ------------------ K2a: per-segment zero-state aggregates tau --------------
// Lane per (c,b,seg): 16384 lanes = 512 waves, 125 dependent steps each.
__global__ void phase2a_segagg(const ChanMats* __restrict__ mats,
                               float* __restrict__ S, float* __restrict__ tau) {
    int id = blockIdx.x * blockDim.x + threadIdx.x;
    if (id >= CH * BATCH * SEG) return;
    int c = id / (BATCH * SEG);
    int s = id % SEG;
    int cb = id / SEG;                 // c*BATCH + b
    float Are[4][4], Aim[4][4];
    load44(mats[c].Are, Are); load44(mats[c].Aim, Aim);
    float sr[4] = {0.f,0.f,0.f,0.f}, si[4] = {0.f,0.f,0.f,0.f};
    float* p = S + ((size_t)cb * NBLK + (size_t)s * SEGBLK) * 8;
    scan_blocks<false>(p, SEGBLK, Are, Aim, sr, si);
    float* t = tau + (size_t)id * 8;
    #pragma unroll
    for (int j = 0; j < 4; ++j) { t[j] = sr[j]; t[4 + j] = si[j]; }
}

// ------------------ K2b: compose segments (16 sequential steps) -------------
// In place on tau: tau[s] becomes the state ENTERING segment s.
__global__ void phase2b_compose(const ChanMats* __restrict__ mats,
                                float* __restrict__ tau) {
    int id = blockIdx.x * blockDim.x + threadIdx.x;
    if (id >= CH * BATCH) return;
    int c = id / BATCH;
    float Sre[4][4], Sim[4][4];
    load44(mats[c].Sre, Sre); load44(mats[c].Sim, Sim);
    float sr[4] = {0.f,0.f,0.f,0.f}, si[4] = {0.f,0.f,0.f,0.f};
    scan_blocks<true>(tau + (size_t)id * SEG * 8, SEG, Sre, Sim, sr, si);
}

// ------------------ K2c: replay segments, write per-block entry states ------
__global__ void phase2c_replay(const ChanMats* __restrict__ mats,
                               float* __restrict__ S,
                               const float* __restrict__ tau) {
    int id = blockIdx.x * blockDim.x + threadIdx.x;
    if (id >= CH * BATCH * SEG) return;
    int c = id / (BATCH * SEG);
    int s = id % SEG;
    int cb = id / SEG;
    float Are[4][4], Aim[4][4];
    load44(mats[c].Are, Are); load44(mats[c].Aim, Aim);
    const float* t = tau + (size_t)id * 8;
    float sr[4], si[4];
    #pragma unroll
    for (int j = 0; j < 4; ++j) { sr[j] = t[j]; si[j] = t[4 + j]; }
    float* p = S + ((size_t)cb * NBLK + (size_t)s * SEGBLK) * 8;
    scan_blocks<true>(p, SEGBLK, Are, Aim, sr, si);
}

// ------------------------------ K3: out = Hr*U + Psi*sigma (WMMA) -----------
__global__ void __launch_bounds__(32)
phase3_output(const float* __restrict__ inp, const ChanMats* __restrict__ mats,
              const float* __restrict__ S, float* __restrict__ out) {
    const int bid  = blockIdx.x;
    const int c    = bid & (CH - 1);
    const int b    = bid >> 7;
    const int lane = threadIdx.x;
    const int half = lane >> 4;
    const int lm   = lane & 15;
    const ChanMats& M = mats[c];

    v2f aH[4];
    #pragma unroll
    for (int q = 0; q < 4; ++q)
        aH[q] = *(const v2f*)&M.Hr[lm][4 * q + 2 * half];
    v2f aPr = *(const v2f*)&M.PsiR[lm][2 * half];
    v2f aPn = *(const v2f*)&M.PsiN[lm][2 * half];

    const float* x    = inp + (size_t)b * TLEN;
    const float* Srow = S + (size_t)(c * BATCH + b) * NBLK * 8;

    for (int tile = 0; tile < NTILE; ++tile) {
        const float* xt = x + tile * (LBLK * TILE);
        v8f acc = {};
        #pragma unroll
        for (int q = 0; q < 4; ++q) {
            v2f bu = *(const v2f*)&xt[lm * LBLK + 4 * q + 2 * half];
            acc = wmma4(aH[q], bu, acc);               // zero-state output (Re)
        }
        // sigma B-frags: K = state index j (0..3), N = block in tile.
        const float* sp = Srow + (size_t)(tile * TILE + lm) * 8;
        v2f bsr = *(const v2f*)&sp[2 * half];
        v2f bsi = *(const v2f*)&sp[4 + 2 * half];
        acc = wmma4(aPr, bsr, acc);                     // + PsiR * sigma_r
        acc = wmma4(aPn, bsi, acc);                     // + (-PsiI) * sigma_i
        // Store out[b][t][c]; t = (tile*16 + N)*16 + M, M = v + 8*half, N = lm.
        int t0 = (tile * TILE + lm) * LBLK + 8 * half;
        float* o = out + ((size_t)b * TLEN + t0) * CH + c;
        #pragma unroll
        for (int v = 0; v < 8; ++v) o[(size_t)v * CH] = acc[v];
    }
}

// ---------------------------------------------------------------------------
extern "C" void kernel_launch(void* const* d_in, const int* in_sizes, int n_in,
                              void* d_out, int out_size, void* d_ws, size_t ws_size,
                              hipStream_t stream) {
    const float* inp     = (const float*)d_in[0];   // [8, 32000]
    const float* coef_re = (const float*)d_in[1];   // [128]
    const float* coef_im = (const float*)d_in[2];   // [128]
    const float* factor  = (const float*)d_in[3];   // [128]
    float* out = (float*)d_out;                     // [8, 32000, 128]

    // Workspace layout: ChanMats[128] | S (CH*B*NBLK*8 f32) | tau (CH*B*SEG*8 f32)
    size_t mats_bytes = (sizeof(ChanMats) * CH + 255) & ~(size_t)255;
    size_t s_bytes    = (size_t)CH * BATCH * NBLK * 8 * sizeof(float);
    ChanMats* mats = (ChanMats*)d_ws;
    float* S   = (float*)((char*)d_ws + mats_bytes);
    float* tau = (float*)((char*)d_ws + mats_bytes + s_bytes);
    (void)in_sizes; (void)n_in; (void)out_size; (void)ws_size;

    precompute_mats<<<1, 128, 0, stream>>>(coef_re, coef_im, factor, mats);
    phase1_states  <<<CH * BATCH, 32, 0, stream>>>(inp, mats, S);
    phase2a_segagg <<<(CH * BATCH * SEG + 255) / 256, 256, 0, stream>>>(mats, S, tau);
    phase2b_compose<<<(CH * BATCH + 127) / 128, 128, 0, stream>>>(mats, tau);
    phase2c_replay <<<(CH * BATCH * SEG + 255) / 256, 256, 0, stream>>>(mats, S, tau);
    phase3_output  <<<CH * BATCH, 32, 0, stream>>>(inp, mats, S, out);
}